// CropModule_79276506349605
// MI455X (gfx1250) — compile-verified
//
#include <hip/hip_runtime.h>
#include <cstdint>

#define IMG_SIZE 480
#define OUT_H 224
#define OUT_W 224
#define BATCH 128
#define LDS_STRIDE 260   // >= max crop width (240), not a multiple of 64 banks

#if defined(__gfx1250__) && __has_builtin(__builtin_amdgcn_global_load_async_to_lds_b32)
#define USE_ASYNC_LDS 1
#else
#define USE_ASYNC_LDS 0
#endif

typedef __attribute__((address_space(1))) int gas_int;  // global (AS1)
typedef __attribute__((address_space(3))) int las_int;  // LDS (AS3)

__global__ __launch_bounds__(256) void crop_resize_kernel(
    const float* __restrict__ img,   // [B, 3, 480, 480]
    const float* __restrict__ box,   // [B, 4] normalized cxcywh
    float* __restrict__ out)         // [B, 3*224*224]
{
    const int oy  = blockIdx.x;      // output row
    const int b   = blockIdx.y;      // batch index
    const int tid = threadIdx.x;

    // 6 staged rows: 3 channels x {gy0, gy1}
    __shared__ float lds[6 * LDS_STRIDE];

    // ---- per-block uniform box math (matches reference exactly) ----
    const float cxf = box[b * 4 + 0] * (float)IMG_SIZE;
    const float cyf = box[b * 4 + 1] * (float)IMG_SIZE;
    const float wf  = box[b * 4 + 2] * (float)IMG_SIZE;
    const float hf  = box[b * 4 + 3] * (float)IMG_SIZE;

    const int xa = (int)truncf(cxf - 0.5f * wf);
    const int ya = (int)truncf(cyf - 0.5f * hf);
    const int xb = (int)truncf(cxf + 0.5f * wf);
    const int yb = (int)truncf(cyf + 0.5f * hf);

    const int   cwI = xb - xa;
    const int   chI = yb - ya;
    const float cw  = (float)cwI;
    const float ch  = (float)chI;

    // vertical sample position for this output row
    float sy = ((float)oy + 0.5f) * ch / (float)OUT_H - 0.5f;
    sy = fminf(fmaxf(sy, 0.0f), ch - 1.0f);
    const int   y0  = (int)floorf(sy);
    const float wyv = sy - (float)y0;
    const int   y1  = min(y0 + 1, chI - 1);
    const int   gy0 = ya + y0;
    const int   gy1 = ya + y1;

    // ---- stage 6 contiguous row-windows [xa, xb) into LDS ----
    const int width = min(cwI, LDS_STRIDE);
    if (tid < width) {  // width <= 240 < 256: one pass covers the row
        #pragma unroll
        for (int chn = 0; chn < 3; ++chn) {
            const long rowBase0 = ((long)(b * 3 + chn) * IMG_SIZE + gy0) * IMG_SIZE + xa + tid;
            const long rowBase1 = ((long)(b * 3 + chn) * IMG_SIZE + gy1) * IMG_SIZE + xa + tid;
            const float* g0 = img + rowBase0;
            const float* g1 = img + rowBase1;
            float* l0 = &lds[(chn * 2 + 0) * LDS_STRIDE + tid];
            float* l1 = &lds[(chn * 2 + 1) * LDS_STRIDE + tid];
#if USE_ASYNC_LDS
            __builtin_amdgcn_global_load_async_to_lds_b32(
                (gas_int*)const_cast<float*>(g0), (las_int*)l0, 0, 0);
            __builtin_amdgcn_global_load_async_to_lds_b32(
                (gas_int*)const_cast<float*>(g1), (las_int*)l1, 0, 0);
#else
            *l0 = *g0;
            *l1 = *g1;
#endif
        }
    }

#if USE_ASYNC_LDS
#if __has_builtin(__builtin_amdgcn_s_wait_asynccnt)
    __builtin_amdgcn_s_wait_asynccnt(0);
#else
    asm volatile("s_wait_asynccnt 0" ::: "memory");
#endif
#endif
    __syncthreads();

    // ---- bilinear blend out of LDS, coalesced store ----
    if (tid < OUT_W) {
        const int ox = tid;
        float sx = ((float)ox + 0.5f) * cw / (float)OUT_W - 0.5f;
        sx = fminf(fmaxf(sx, 0.0f), cw - 1.0f);
        const int   x0  = (int)floorf(sx);
        const float wxv = sx - (float)x0;
        const int   x1  = min(x0 + 1, cwI - 1);
        const float omx = 1.0f - wxv;
        const float omy = 1.0f - wyv;

        #pragma unroll
        for (int chn = 0; chn < 3; ++chn) {
            const float* r0 = &lds[(chn * 2 + 0) * LDS_STRIDE];
            const float* r1 = &lds[(chn * 2 + 1) * LDS_STRIDE];
            const float top = r0[x0] * omx + r0[x1] * wxv;
            const float bot = r1[x0] * omx + r1[x1] * wxv;
            const float v   = top * omy + bot * wyv;
            out[(((long)b * 3 + chn) * OUT_H + oy) * OUT_W + ox] = v;
        }
    }
}

extern "C" void kernel_launch(void* const* d_in, const int* in_sizes, int n_in,
                              void* d_out, int out_size, void* d_ws, size_t ws_size,
                              hipStream_t stream) {
    const float* img = (const float*)d_in[0];  // [128,3,480,480] f32
    const float* box = (const float*)d_in[1];  // [128,4] f32
    float* out = (float*)d_out;                // [128, 3*224*224] f32
    (void)in_sizes; (void)n_in; (void)out_size; (void)d_ws; (void)ws_size;

    dim3 grid(OUT_H, BATCH, 1);   // one output row per block
    dim3 block(256, 1, 1);        // 8 wave32 waves
    crop_resize_kernel<<<grid, block, 0, stream>>>(img, box, out);
}